// DifferentiableNeuralComputer_58325655879880
// MI455X (gfx1250) — compile-verified
//
#include <hip/hip_runtime.h>
#include <math.h>

// ---------------- problem constants ----------------
#define BB   16
#define NN   2048
#define WD   64
#define RR   4
#define INH  256
#define HH   512
#define IFACE 727   // R*Wd + R + Wd + 1 + Wd + Wd + R + 1 + 1 + 3R + 256

// ---------------- workspace layout (floats) ----------------
constexpr int H_OFF = 0;                       // h: 16x512
constexpr int V_OFF = H_OFF + BB * HH;         // v: 16x727
constexpr int RKN   = V_OFF + BB * IFACE;      // normalized read keys: B*R*64
constexpr int WKN   = RKN + BB * RR * WD;      // normalized write key:  B*64
constexpr int ERS   = WKN + BB * WD;           // erase vector sigmoid:  B*64
constexpr int WVV   = ERS + BB * WD;           // write vector:          B*64
constexpr int SCAL  = WVV + BB * WD;           // per-b: wg, ag, u0, pad : B*4
constexpr int RM    = SCAL + BB * 4;           // read modes softmaxed:  B*R*3
constexpr int SW    = RM + BB * RR * 3;        // write content scores:  B*N
constexpr int WW    = SW + BB * NN;            // write weights:         B*N
constexpr int SUMW  = WW + BB * NN;            // sum of write weights:  B
constexpr int PAD0  = SUMW + BB;               // pad so ROWL is 16B aligned
constexpr int ROWL  = (PAD0 + 3) & ~3;         // rowsum(L):   B*N   (atomic)
constexpr int LWV   = ROWL + BB * NN;          // L·w:         B*N   (atomic)
constexpr int COLL  = LWV + BB * NN;           // colsum(L):   B*N   (atomic)
constexpr int WTL   = COLL + BB * NN;          // wᵀ·L:        B*N   (atomic)
constexpr int FWD   = WTL + BB * NN;           // fwd read weight (per b): B*N
constexpr int BWD   = FWD + BB * NN;           // bwd read weight (per b): B*N
constexpr int SCR   = BWD + BB * NN;           // read content scores: B*R*N

#define EPSF 1e-8f

typedef float v2f __attribute__((ext_vector_type(2)));
typedef float v4f __attribute__((ext_vector_type(4)));
typedef float v8f __attribute__((ext_vector_type(8)));
typedef int   v4i_ __attribute__((__vector_size__(4 * sizeof(int))));

// gfx1250 async global->LDS path (ASYNCcnt), guarded so it can't break compile
#if defined(__HIP_DEVICE_COMPILE__) && \
    __has_builtin(__builtin_amdgcn_global_load_async_to_lds_b128) && \
    __has_builtin(__builtin_amdgcn_s_wait_asynccnt)
#define USE_ASYNC_LDS 1
#endif

// ---------------- helpers ----------------
__device__ __forceinline__ float sigmoidf_(float x) { return 1.f / (1.f + __expf(-x)); }
__device__ __forceinline__ float softplusf_(float x) { return (x > 20.f) ? x : log1pf(__expf(x)); }

__device__ __forceinline__ float wave_sum(float v) {
    #pragma unroll
    for (int o = 16; o > 0; o >>= 1) v += __shfl_xor(v, o, 32);
    return v;
}
__device__ __forceinline__ float wave_max(float v) {
    #pragma unroll
    for (int o = 16; o > 0; o >>= 1) v = fmaxf(v, __shfl_xor(v, o, 32));
    return v;
}
// 256-thread block reduce; sh must have 8 floats
__device__ __forceinline__ float block_sum(float v, float* sh, int tid) {
    v = wave_sum(v);
    if ((tid & 31) == 0) sh[tid >> 5] = v;
    __syncthreads();
    float r = sh[0];
    #pragma unroll
    for (int i = 1; i < 8; ++i) r += sh[i];
    __syncthreads();
    return r;
}
__device__ __forceinline__ float block_max(float v, float* sh, int tid) {
    v = wave_max(v);
    if ((tid & 31) == 0) sh[tid >> 5] = v;
    __syncthreads();
    float r = sh[0];
    #pragma unroll
    for (int i = 1; i < 8; ++i) r = fmaxf(r, sh[i]);
    __syncthreads();
    return r;
}

// ---------------- kernel 0: zero atomic accumulators (b128 stores) -----------
__global__ void dnc_zero(float* ws) {
    int i = blockIdx.x * 256 + threadIdx.x;   // covers 4*B*N floats at ROWL, /4
    v4f z = {};
    ((v4f*)(ws + ROWL))[i] = z;
}

// ---------------- kernel 1: h = tanh(x@W1 + b1), WMMA f32 16x16x4 ------------
__global__ void dnc_gemm1(const float* __restrict__ x, const float* __restrict__ W1,
                          const float* __restrict__ b1, float* __restrict__ ws) {
    int lane = threadIdx.x;          // 32 threads
    int n0   = blockIdx.x * 16;      // 32 tiles over H=512
    int m    = lane & 15;            // row (A) / col (B,D)
    int kh   = lane >> 4;            // K-half select
    v8f acc = {};
    for (int k0 = 0; k0 < INH; k0 += 4) {
        int ka = k0 + 2 * kh;
        v2f a, b;
        a.x = x[m * INH + ka];
        a.y = x[m * INH + ka + 1];
        b.x = W1[(size_t)ka * HH + n0 + m];
        b.y = W1[(size_t)(ka + 1) * HH + n0 + m];
        acc = __builtin_amdgcn_wmma_f32_16x16x4_f32(false, a, false, b,
                                                    (short)0, acc, false, false);
    }
    #pragma unroll
    for (int j = 0; j < 8; ++j) {
        int row = j + 8 * kh;
        int col = n0 + m;
        ws[H_OFF + row * HH + col] = tanhf(acc[j] + b1[col]);
    }
}

// ---------------- kernel 2: v = h@W2 + b2, WMMA f32 16x16x4 ------------------
__global__ void dnc_gemm2(const float* __restrict__ W2, const float* __restrict__ b2,
                          float* __restrict__ ws) {
    int lane = threadIdx.x;
    int n0   = blockIdx.x * 16;      // 46 tiles over 727 (last partial)
    int m    = lane & 15;
    int kh   = lane >> 4;
    int col  = n0 + m;
    v8f acc = {};
    for (int k0 = 0; k0 < HH; k0 += 4) {
        int ka = k0 + 2 * kh;
        v2f a, b;
        a.x = ws[H_OFF + m * HH + ka];
        a.y = ws[H_OFF + m * HH + ka + 1];
        b.x = (col < IFACE) ? W2[(size_t)ka * IFACE + col] : 0.f;
        b.y = (col < IFACE) ? W2[(size_t)(ka + 1) * IFACE + col] : 0.f;
        acc = __builtin_amdgcn_wmma_f32_16x16x4_f32(false, a, false, b,
                                                    (short)0, acc, false, false);
    }
    if (col < IFACE) {
        #pragma unroll
        for (int j = 0; j < 8; ++j) {
            int row = j + 8 * kh;
            ws[V_OFF + row * IFACE + col] = acc[j] + b2[col];
        }
    }
}

// ---------------- kernel 3: interface activations (one block per batch) ------
__global__ void dnc_iface(float* __restrict__ ws) {
    int b = blockIdx.x, tid = threadIdx.x;   // 256 threads
    const float* vb = ws + V_OFF + b * IFACE;
    __shared__ float sk[5][WD];
    __shared__ float snrm[5];
    {   // scaled read keys (strength folded in before normalization, as in ref)
        int r = tid >> 6, w = tid & 63;
        float s = 1.f + softplusf_(vb[256 + r]);
        sk[r][w] = vb[r * WD + w] * s;
    }
    if (tid < WD) {   // scaled write key
        float s = 1.f + softplusf_(vb[324]);
        sk[4][tid] = vb[260 + tid] * s;
    }
    __syncthreads();
    if (tid < 5) {
        float acc = 0.f;
        for (int w = 0; w < WD; ++w) { float t = sk[tid][w]; acc += t * t; }
        snrm[tid] = sqrtf(acc) + EPSF;
    }
    __syncthreads();
    {   // normalized read keys
        int r = tid >> 6, w = tid & 63;
        ws[RKN + (b * RR + r) * WD + w] = sk[r][w] / snrm[r];
    }
    if (tid < WD) {
        ws[WKN + b * WD + tid] = sk[4][tid] / snrm[4];
        ws[ERS + b * WD + tid] = sigmoidf_(vb[325 + tid]);
        ws[WVV + b * WD + tid] = vb[389 + tid];
    }
    if (tid == 0) {
        // var_phi is constant over n -> argsort is identity -> closed-form allocation
        float c = 1.f;
        #pragma unroll
        for (int r = 0; r < RR; ++r)
            c *= (1.f - sigmoidf_(vb[453 + r]) * (1.f / (float)NN));
        ws[SCAL + b * 4 + 0] = sigmoidf_(vb[458]);      // write gate
        ws[SCAL + b * 4 + 1] = sigmoidf_(vb[457]);      // allocation gate
        ws[SCAL + b * 4 + 2] = 1e-4f * c;               // u0
    }
    if (tid < RR) {   // read-mode softmax over 3
        float m0 = vb[459 + tid * 3], m1 = vb[459 + tid * 3 + 1], m2 = vb[459 + tid * 3 + 2];
        float mx = fmaxf(m0, fmaxf(m1, m2));
        float e0 = __expf(m0 - mx), e1 = __expf(m1 - mx), e2 = __expf(m2 - mx);
        float inv = 1.f / (e0 + e1 + e2);
        ws[RM + (b * RR + tid) * 3 + 0] = e0 * inv;
        ws[RM + (b * RR + tid) * 3 + 1] = e1 * inv;
        ws[RM + (b * RR + tid) * 3 + 2] = e2 * inv;
    }
}

// ---------------- kernel 4: write content scores (one wave per (b,n)) --------
__global__ void dnc_wscore(const float* __restrict__ M, float* __restrict__ ws) {
    int gw = blockIdx.x * 8 + (threadIdx.x >> 5);
    int lane = threadIdx.x & 31;
    int b = gw >> 11, n = gw & (NN - 1);
    const float* m = M + (size_t)(b * NN + n) * WD;
    v2f mv = *(const v2f*)(m + lane * 2);                         // b64 load
    v2f kv = *(const v2f*)(ws + WKN + b * WD + lane * 2);
    float nr = wave_sum(mv.x * mv.x + mv.y * mv.y);
    float dt = wave_sum(mv.x * kv.x + mv.y * kv.y);
    if (lane == 0) ws[SW + b * NN + n] = dt / (sqrtf(nr) + EPSF);
}

// ---------------- kernel 5: write softmax + write weights + sumW -------------
__global__ void dnc_wsoftmax(float* __restrict__ ws) {
    int b = blockIdx.x, tid = threadIdx.x;   // 256
    __shared__ float red[8];
    const float* s = ws + SW + b * NN;
    float lmax = -3.4e38f;
    for (int i = tid; i < NN; i += 256) lmax = fmaxf(lmax, s[i]);
    lmax = block_max(lmax, red, tid);
    float lsum = 0.f;
    for (int i = tid; i < NN; i += 256) lsum += __expf(s[i] - lmax);
    float sum = block_sum(lsum, red, tid);
    float inv = 1.f / sum;
    float wg = ws[SCAL + b * 4 + 0], ag = ws[SCAL + b * 4 + 1], u0 = ws[SCAL + b * 4 + 2];
    float lnu0 = __logf(u0), om = 1.f - u0;
    float lw = 0.f;
    for (int i = tid; i < NN; i += 256) {
        float cw = __expf(s[i] - lmax) * inv;                 // content write weight
        float alloc = om * __expf((float)(i + 1) * lnu0);     // closed-form allocation
        float w = wg * (ag * alloc + (1.f - ag) * cw);
        ws[WW + b * NN + i] = w;
        lw += w;
    }
    float sw = block_sum(lw, red, tid);
    if (tid == 0) ws[SUMW + b] = sw;
}

// ---------------- kernel 6: THE big one — single pass over L (256 MB) --------
// rowsum(L), L·w, colsum(L), wᵀ·L with one NT b128-streamed read of L.
// Tile: 128 rows x 256 cols per block; 8 waves x 16 rows; 2 float4 cols/lane.
__global__ void dnc_lstream(const float* __restrict__ L, float* __restrict__ ws) {
    int b = blockIdx.z;
    int rowBase = blockIdx.y * 128;
    int colBase = blockIdx.x * 256;
    int tid = threadIdx.x, wave = tid >> 5, lane = tid & 31;
    __shared__ float wrow[128];
    __shared__ float cacc[256];
    __shared__ float cwacc[256];
    cacc[tid]  = 0.f;
    cwacc[tid] = 0.f;
#ifdef USE_ASYNC_LDS
    // stage this block's 128 row write-weights via async global->LDS (ASYNCcnt)
    if (tid < 32) {
        __builtin_amdgcn_global_load_async_to_lds_b128(
            (__attribute__((address_space(1))) v4i_*)(ws + WW + b * NN + rowBase + tid * 4),
            (__attribute__((address_space(3))) v4i_*)(&wrow[tid * 4]), 0, 0);
        __builtin_amdgcn_s_wait_asynccnt(0);
    }
#else
    if (tid < 128) wrow[tid] = ws[WW + b * NN + rowBase + tid];
#endif
    // loop-invariant column write-weights: 8 cols per lane, kept in registers
    v4f wc0 = *(const v4f*)(ws + WW + b * NN + colBase + lane * 4);
    v4f wc1 = *(const v4f*)(ws + WW + b * NN + colBase + 128 + lane * 4);
    __syncthreads();

    v4f ca0 = {}, ca1 = {}, cwa0 = {}, cwa1 = {};
    const float* Lb = L + (size_t)b * NN * NN;

    for (int i = 0; i < 16; ++i) {
        int r = rowBase + wave * 16 + i;
        float wr = wrow[wave * 16 + i];
        const float* Lr = Lb + (size_t)r * NN + colBase;
        // speculative prefetch of the next row segment (gfx1250 global_prefetch_b8)
        __builtin_prefetch(Lr + NN + lane * 32, 0, 0);
        // two NT b128 loads per lane per row: 1 KB per wave-row, streams past L2
        v4f l0 = __builtin_nontemporal_load((const v4f*)(Lr + lane * 4));
        v4f l1 = __builtin_nontemporal_load((const v4f*)(Lr + 128 + lane * 4));
        float rowsum = (l0.x + l0.y + l0.z + l0.w) + (l1.x + l1.y + l1.z + l1.w);
        float rdot   = l0.x * wc0.x + l0.y * wc0.y + l0.z * wc0.z + l0.w * wc0.w
                     + l1.x * wc1.x + l1.y * wc1.y + l1.z * wc1.z + l1.w * wc1.w;
        ca0 += l0;  ca1 += l1;
        cwa0 += wr * l0;  cwa1 += wr * l1;
        rowsum = wave_sum(rowsum);
        rdot   = wave_sum(rdot);
        if (lane == 0) {
            atomicAdd(&ws[ROWL + b * NN + r], rowsum);
            atomicAdd(&ws[LWV  + b * NN + r], rdot);
        }
    }
    // combine per-lane column partials across the 8 waves in LDS (ds_add_f32)
    #pragma unroll
    for (int k = 0; k < 4; ++k) {
        atomicAdd(&cacc[lane * 4 + k],        ca0[k]);
        atomicAdd(&cacc[128 + lane * 4 + k],  ca1[k]);
        atomicAdd(&cwacc[lane * 4 + k],       cwa0[k]);
        atomicAdd(&cwacc[128 + lane * 4 + k], cwa1[k]);
    }
    __syncthreads();
    atomicAdd(&ws[COLL + b * NN + colBase + tid], cacc[tid]);
    atomicAdd(&ws[WTL  + b * NN + colBase + tid], cwacc[tid]);
}

// ---------------- kernel 7: RS/CS from L reductions, softmax -> bwd/fwd ------
__global__ void dnc_rscs(const float* __restrict__ p, float* __restrict__ ws) {
    int b = blockIdx.x, tid = threadIdx.x;   // 256
    __shared__ float rsb[NN];
    __shared__ float csb[NN];
    __shared__ float red[8];
    float lp = 0.f;
    for (int i = tid; i < NN; i += 256) lp += p[b * NN + i];
    float sumP = block_sum(lp, red, tid);
    float sumW = ws[SUMW + b];
    const float invN = 1.f / (float)NN;
    for (int i = tid; i < NN; i += 256) {
        float w  = ws[WW   + b * NN + i];
        float pi = p[b * NN + i];
        // RS[n] = (1-w_n)*rowL - (Lw)_n + w_n*(sumP - p_n)   (L diag already 0)
        float rs = (1.f - w) * ws[ROWL + b * NN + i] - ws[LWV + b * NN + i] + w * (sumP - pi);
        // CS[m] = (1-w_m)*colL - (wᵀL)_m + p_m*(sumW - w_m)
        float cs = (1.f - w) * ws[COLL + b * NN + i] - ws[WTL + b * NN + i] + pi * (sumW - w);
        rsb[i] = rs * invN;   // prev_rw is uniform 1/N
        csb[i] = cs * invN;
    }
    __syncthreads();
    float mr = -3.4e38f, mc = -3.4e38f;
    for (int i = tid; i < NN; i += 256) { mr = fmaxf(mr, rsb[i]); mc = fmaxf(mc, csb[i]); }
    mr = block_max(mr, red, tid);
    mc = block_max(mc, red, tid);
    float sr = 0.f, sc = 0.f;
    for (int i = tid; i < NN; i += 256) { sr += __expf(rsb[i] - mr); sc += __expf(csb[i] - mc); }
    sr = block_sum(sr, red, tid);
    sc = block_sum(sc, red, tid);
    float isr = 1.f / sr, isc = 1.f / sc;
    for (int i = tid; i < NN; i += 256) {
        ws[BWD + b * NN + i] = __expf(rsb[i] - mr) * isr;
        ws[FWD + b * NN + i] = __expf(csb[i] - mc) * isc;
    }
}

// ---------------- kernel 8: read content scores (memory_new recomputed) ------
__global__ void dnc_rscore(const float* __restrict__ M, float* __restrict__ ws) {
    int gw = blockIdx.x * 8 + (threadIdx.x >> 5);
    int lane = threadIdx.x & 31;
    int b = gw >> 11, n = gw & (NN - 1);
    const float* m = M + (size_t)(b * NN + n) * WD;
    float wn = ws[WW + b * NN + n];
    v2f mv = *(const v2f*)(m + lane * 2);
    v2f ev = *(const v2f*)(ws + ERS + b * WD + lane * 2);
    v2f vv = *(const v2f*)(ws + WVV + b * WD + lane * 2);
    float mn0 = mv.x * (1.f - wn * ev.x) + wn * vv.x;
    float mn1 = mv.y * (1.f - wn * ev.y) + wn * vv.y;
    float nr = wave_sum(mn0 * mn0 + mn1 * mn1);
    float d[RR];
    #pragma unroll
    for (int r = 0; r < RR; ++r) {
        v2f kv = *(const v2f*)(ws + RKN + (b * RR + r) * WD + lane * 2);
        d[r] = wave_sum(mn0 * kv.x + mn1 * kv.y);
    }
    if (lane == 0) {
        float inv = 1.f / (sqrtf(nr) + EPSF);
        #pragma unroll
        for (int r = 0; r < RR; ++r)
            ws[SCR + (b * RR + r) * NN + n] = d[r] * inv;
    }
}

// ---------------- kernel 9: read content softmax (per (b,r), in place) -------
__global__ void dnc_rsoftmax(float* __restrict__ ws) {
    int br = blockIdx.x, tid = threadIdx.x;   // 64 blocks x 256
    float* s = ws + SCR + br * NN;
    __shared__ float red[8];
    float lmax = -3.4e38f;
    for (int i = tid; i < NN; i += 256) lmax = fmaxf(lmax, s[i]);
    lmax = block_max(lmax, red, tid);
    float ls = 0.f;
    for (int i = tid; i < NN; i += 256) ls += __expf(s[i] - lmax);
    float sum = block_sum(ls, red, tid);
    float inv = 1.f / sum;
    for (int i = tid; i < NN; i += 256) s[i] = __expf(s[i] - lmax) * inv;
}

// ---------------- kernel 10: read weights + read vectors -> output -----------
__global__ void dnc_readvec(const float* __restrict__ M, const float* __restrict__ ws,
                            float* __restrict__ out) {
    int b = blockIdx.x, tid = threadIdx.x;   // 256: tid -> (r, w)
    int r = tid >> 6, w = tid & 63;
    float rm0 = ws[RM + (b * RR + r) * 3 + 0];
    float rm1 = ws[RM + (b * RR + r) * 3 + 1];
    float rm2 = ws[RM + (b * RR + r) * 3 + 2];
    float ev = ws[ERS + b * WD + w];
    float vv = ws[WVV + b * WD + w];
    const float* bwd  = ws + BWD + b * NN;
    const float* fwd  = ws + FWD + b * NN;
    const float* cont = ws + SCR + (b * RR + r) * NN;
    const float* wwp  = ws + WW  + b * NN;
    const float* Mb   = M + (size_t)b * NN * WD;
    float acc = 0.f;
    for (int n = 0; n < NN; ++n) {
        float rw = rm0 * bwd[n] + rm1 * cont[n] + rm2 * fwd[n];
        float wn = wwp[n];
        float mn = Mb[(size_t)n * WD + w] * (1.f - wn * ev) + wn * vv;  // memory_new
        acc += rw * mn;
    }
    out[b * (RR * WD) + tid] = acc;
}

// ---------------- launcher ----------------
extern "C" void kernel_launch(void* const* d_in, const int* in_sizes, int n_in,
                              void* d_out, int out_size, void* d_ws, size_t ws_size,
                              hipStream_t stream) {
    const float* x   = (const float*)d_in[0];
    const float* M   = (const float*)d_in[1];
    const float* L   = (const float*)d_in[2];
    const float* p   = (const float*)d_in[3];
    const float* W1  = (const float*)d_in[4];
    const float* b1  = (const float*)d_in[5];
    const float* W2  = (const float*)d_in[6];
    const float* b2  = (const float*)d_in[7];
    float* out = (float*)d_out;
    float* ws  = (float*)d_ws;

    dnc_zero    <<<(4 * BB * NN / 4) / 256, 256, 0, stream>>>(ws);
    dnc_gemm1   <<<HH / 16, 32, 0, stream>>>(x, W1, b1, ws);
    dnc_gemm2   <<<(IFACE + 15) / 16, 32, 0, stream>>>(W2, b2, ws);
    dnc_iface   <<<BB, 256, 0, stream>>>(ws);
    dnc_wscore  <<<(BB * NN) / 8, 256, 0, stream>>>(M, ws);
    dnc_wsoftmax<<<BB, 256, 0, stream>>>(ws);
    dnc_lstream <<<dim3(NN / 256, NN / 128, BB), 256, 0, stream>>>(L, ws);
    dnc_rscs    <<<BB, 256, 0, stream>>>(p, ws);
    dnc_rscore  <<<(BB * NN) / 8, 256, 0, stream>>>(M, ws);
    dnc_rsoftmax<<<BB * RR, 256, 0, stream>>>(ws);
    dnc_readvec <<<BB, 256, 0, stream>>>(M, ws, out);
}